// MultiFreqEncoder_2671469658639
// MI455X (gfx1250) — compile-verified
//
#include <hip/hip_runtime.h>
#include <hip/hip_bf16.h>
#include <stdint.h>

// MultiFreqEncoder for MI455X (gfx1250, wave32).
// Bandwidth-bound: 604 MB of streamed output dominates. Tables (48.4 KB) are
// staged in LDS via CDNA5 async global->LDS copies; per-point sinusoid/index
// params are staged in LDS so phase 2 can emit fully coalesced b128 NT stores.

#define XD        3
#define NFREQ     6
#define NCH       8
#define COMBOS    36              // NFREQ * 2 * XD
#define PBLK      256             // points per block
#define TPB       256             // threads per block (8 waves, wave32)
#define TAB_FLOATS 12096          // sum over s of 48 * (4<<s)

typedef __attribute__((ext_vector_type(4))) float v4f;
// Exact pointee type expected by the async-load builtin (per clang diagnostic):
typedef int v4i_gcc __attribute__((__vector_size__(16)));
typedef __attribute__((address_space(1))) v4i_gcc* gptr_v4i;   // global (__device__)
typedef __attribute__((address_space(3))) v4i_gcc* sptr_v4i;   // LDS   (__shared__)

// ---- CDNA5 async global -> LDS copy (ASYNCcnt path), guarded -------------
__device__ __forceinline__ void async_cp_f4(const float* g, float* l) {
#if __has_builtin(__builtin_amdgcn_global_load_async_to_lds_b128)
  // Cast through integers: inttoptr into AS1/AS3 is always legal.
  // Flat addr == global addr for global memory; flat addr[31:0] == LDS offset.
  __builtin_amdgcn_global_load_async_to_lds_b128(
      (gptr_v4i)(uintptr_t)g,
      (sptr_v4i)(uint32_t)(uintptr_t)l,
      /*offset=*/0, /*cpol=*/0);
#else
  // Fallback: synchronous copy (vmem load + ds store)
  *(v4f*)l = *(const v4f*)g;
#endif
}

__device__ __forceinline__ void wait_async_all() {
#if __has_builtin(__builtin_amdgcn_s_wait_asynccnt)
  __builtin_amdgcn_s_wait_asynccnt(0);
#elif __has_builtin(__builtin_amdgcn_global_load_async_to_lds_b128)
  asm volatile("s_wait_asynccnt 0" ::: "memory");
#endif
}

__global__ __launch_bounds__(TPB)
void mfe_kernel(const float* __restrict__ x,
                const float* __restrict__ f0, const float* __restrict__ f1,
                const float* __restrict__ f2, const float* __restrict__ f3,
                const float* __restrict__ f4, const float* __restrict__ f5,
                float* __restrict__ out, int npts)
{
  // Tables kept in input layout [sd(6)][c(8)][H] per scale, concatenated.
  __shared__ float  sTab[TAB_FLOATS];                 // 48384 B
  __shared__ float2 sPrm[PBLK * COMBOS];              // 73728 B: {lat, pack(y0,w)}

  const int tid = threadIdx.x;

  // ---- stage all feature tables into LDS (async, CDNA5) ------------------
  {
    const float* ft[NFREQ] = {f0, f1, f2, f3, f4, f5};
    int off = 0;
#pragma unroll
    for (int t = 0; t < NFREQ; ++t) {
      const int nf4 = (48 * (4 << t)) / 4;            // float4 count, all /4 exact
      for (int i = tid; i < nf4; i += TPB)
        async_cp_f4(ft[t] + i * 4, &sTab[off + i * 4]);
      off += nf4 * 4;
    }
  }
  wait_async_all();
  __syncthreads();

  const int n0 = blockIdx.x * PBLK;
  const int pc = min(PBLK, npts - n0);

  // ---- phase 1: one thread per point; 36 combos, each computed once ------
  if (tid < pc) {
    const int p = n0 + tid;
    float xv[3];
    xv[0] = x[p * 3 + 0];
    xv[1] = x[p * 3 + 1];
    xv[2] = x[p * 3 + 2];
#pragma unroll
    for (int s = 0; s < NFREQ; ++s) {
      const int   H  = 4 << s;
      const float sc = (float)(1 << s);
#pragma unroll
      for (int d = 0; d < XD; ++d) {
        const float xb = xv[d] * sc;
        const float l0 = __sinf(xb);                  // sin(xb)
        const float l1 = __cosf(xb);                  // sin(xb + pi/2)
#pragma unroll
        for (int ss = 0; ss < 2; ++ss) {
          const float lat = ss ? l1 : l0;
          float y = (lat + 1.0f) * (0.5f * (float)(H - 1));
          y = fminf(fmaxf(y, 0.0f), (float)(H - 1));
          int y0 = (int)y;                            // y >= 0: trunc == floor
          y0 = min(y0, H - 2);
          const float w = y - (float)y0;
          // pack: y0 in [31:24], w as 24-bit fixed point in [23:0]
          const uint32_t wq = (uint32_t)fminf(w * 16777216.0f, 16777215.0f);
          const uint32_t pk = ((uint32_t)y0 << 24) | wq;
          sPrm[tid * COMBOS + (s * 6 + ss * 3 + d)] =
              make_float2(lat, __uint_as_float(pk));
        }
      }
    }
  }
  __syncthreads();

  // ---- phase 2: coalesced b128 NT stores; 72 quads per point -------------
  // out offset within point = c*36 + k, k = s*6 + ss*3 + d.
  // 36 % 4 == 0  =>  every float4 quad has a single c, k in {k0..k0+3}.
  const int nquad = pc * 72;
  v4f* out4 = (v4f*)out + (size_t)n0 * 72;
  for (int q = tid; q < nquad; q += TPB) {
    const int o   = q * 4;
    const int p   = o / 288;
    const int rem = o - p * 288;
    const int c   = rem / 36;
    const int k0  = rem - c * 36;
    float r[4];
#pragma unroll
    for (int j = 0; j < 4; ++j) {
      const int k = k0 + j;
      const float2 prm = sPrm[p * COMBOS + k];
      const uint32_t pk = __float_as_uint(prm.y);
      const int   y0 = (int)(pk >> 24);
      const float w  = (float)(pk & 0xFFFFFFu) * (1.0f / 16777216.0f);
      const int s    = k / 6;
      const int sd   = k - s * 6;
      const int H    = 4 << s;
      const int toff = 192 * ((1 << s) - 1);          // closed-form table base
      const int idx  = toff + (sd * NCH + c) * H + y0;
      const float v0 = sTab[idx];
      const float v1 = sTab[idx + 1];
      r[j] = fmaf(w, v1 - v0, v0) + prm.x;
    }
    v4f rv = {r[0], r[1], r[2], r[3]};
    __builtin_nontemporal_store(rv, &out4[q]);        // streamed once, NT
  }
}

extern "C" void kernel_launch(void* const* d_in, const int* in_sizes, int n_in,
                              void* d_out, int out_size, void* d_ws, size_t ws_size,
                              hipStream_t stream) {
  (void)n_in; (void)out_size; (void)d_ws; (void)ws_size;
  const float* x  = (const float*)d_in[0];
  const float* f0 = (const float*)d_in[1];
  const float* f1 = (const float*)d_in[2];
  const float* f2 = (const float*)d_in[3];
  const float* f3 = (const float*)d_in[4];
  const float* f4 = (const float*)d_in[5];
  const float* f5 = (const float*)d_in[6];
  float* out = (float*)d_out;
  const int npts   = in_sizes[0] / XD;
  const int blocks = (npts + PBLK - 1) / PBLK;
  hipLaunchKernelGGL(mfe_kernel, dim3(blocks), dim3(TPB), 0, stream,
                     x, f0, f1, f2, f3, f4, f5, out, npts);
}